// SpaceTimeStepLookTable_45079976738869
// MI455X (gfx1250) — compile-verified
//
#include <hip/hip_runtime.h>

// SpaceTimeStepLookTable fused gather+ReLU+GEMM for gfx1250 (MI455X).
// Tall-skinny GEMM M=2048, K=37848, N=128 via V_WMMA_F32_16X16X4_F32.
// HBM-gather bound (~310 MB). To keep W_mlp L2 traffic sane:
//   - M-tile = 64 per WG (4 sub-tiles reuse each B chunk in registers)
//   - K-split = 4 across WGs (588 = 4*147 chunks) -> 32x4 = 128 WGs
//   - partial [N,128] sums in workspace, combined by a tiny second kernel.

typedef float v2f __attribute__((ext_vector_type(2)));
typedef float v8f __attribute__((ext_vector_type(8)));

#define KTOT      37848
#define MT        64        // points per workgroup (4 WMMA sub-tiles)
#define SA_STRIDE 66        // 64-float A rows padded (8B aligned, bank-rotated)
#define NB_STRIDE 220       // 216 neighbor feats padded (16B-aligned stride)
#define NCHUNK    588       // (64+512+4096+32768+64+64+64)/64
#define NSPLIT    4
#define CPS       147       // chunks per K-split (588/4)

__device__ __forceinline__ void chunk_map(int c, int& seg, int& kseg, int& kglob) {
    // seg chunk counts: 1, 8, 64, 512, 1, 1, 1 ; cum: 1,9,73,585,586,587,588
    int kc;
    if      (c <   1) { seg = 0; kc = c;       kglob = 0;     }
    else if (c <   9) { seg = 1; kc = c - 1;   kglob = 64;    }
    else if (c <  73) { seg = 2; kc = c - 9;   kglob = 576;   }
    else if (c < 585) { seg = 3; kc = c - 73;  kglob = 4672;  }
    else if (c < 586) { seg = 4; kc = 0;       kglob = 37440; }
    else if (c < 587) { seg = 5; kc = 0;       kglob = 37504; }
    else              { seg = 6; kc = 0;       kglob = 37568; }
    kseg  = kc * 64;
    kglob += kseg;
}

__global__ __launch_bounds__(256)
void stslt_gemm_kernel(const float* __restrict__ pos,  const float* __restrict__ tarr,
                       const float* __restrict__ table0, const float* __restrict__ table1,
                       const float* __restrict__ table2, const float* __restrict__ table3,
                       const float* __restrict__ ts1,    const float* __restrict__ ts2,
                       const float* __restrict__ Wmlp,   float* __restrict__ P,
                       int nPts)
{
    __shared__ const float* sPtr[MT][7];      // per-point row base ptr per segment
    __shared__ int   sI4[MT][4];              // idx3.xyz, t_idx (neighbor taps)
    __shared__ float sNb[MT * NB_STRIDE];     // 54x4 neighbor feats, relu'd
    __shared__ float sA[2][MT * SA_STRIDE];   // double-buffered 64x64 A tiles

    const int tid   = threadIdx.x;
    const int lane  = tid & 31;
    const int wave  = tid >> 5;
    const int p0    = blockIdx.x * MT;
    const int split = blockIdx.y;             // K-split id, uniform per block

    // ---- setup: per-point table row bases ----
    if (tid < MT) {
        const int m = tid, p = p0 + m;
        const float px = pos[p*3+0], py = pos[p*3+1], pz = pos[p*3+2];
        const float tv = tarr[p];
        const int i0x = (int)(px*127.f), i0y = (int)(py*127.f), i0z = (int)(pz*127.f);
        const int i1x = (int)(px*63.f),  i1y = (int)(py*63.f),  i1z = (int)(pz*63.f);
        const int i2x = (int)(px*31.f),  i2y = (int)(py*31.f),  i2z = (int)(pz*31.f);
        const int i3x = (int)(px*15.f),  i3y = (int)(py*15.f),  i3z = (int)(pz*15.f);
        const int ti  = (int)(tv*127.f);
        sPtr[m][0] = table0 + ((long)((i0x*128 + i0y)*128 + i0z)) * 64;
        sPtr[m][1] = table1 + ((long)((i1x*64  + i1y)*64  + i1z)) * 512;
        sPtr[m][2] = table2 + ((long)((i2x*32  + i2y)*32  + i2z)) * 4096;
        sPtr[m][3] = table3 + ((long)((i3x*16  + i3y)*16  + i3z)) * 32768;
        const long tb = ((long)((i3x*16 + i3y)*16 + i3z)) * 4096;   // *64*64
        sPtr[m][4] = ts1 + tb + (long)((ti + 63) % 64) * 64;        // delta -1
        sPtr[m][5] = ts1 + tb + (long)( ti        % 64) * 64;       // delta  0
        sPtr[m][6] = ts1 + tb + (long)((ti + 1)  % 64) * 64;        // delta +1
        sI4[m][0] = i3x; sI4[m][1] = i3y; sI4[m][2] = i3z; sI4[m][3] = ti;
    }
    __syncthreads();

    // ---- split 3 only: gather 54 neighbor taps (4 floats) per point ----
    if (split == NSPLIT - 1) {
        for (int task = tid; task < MT*54; task += 256) {
            const int m = task / 54, j = task % 54;
            const int dt = (j & 1) ? 1 : -1;          // dt in (-1, 1), innermost
            const int dz = ((j >> 1) % 3) - 1;
            const int dy = ((j / 6)  % 3) - 1;
            const int dx =  (j / 18)      - 1;
            const int x  = (sI4[m][0] + dx + 128) & 127;
            const int y  = (sI4[m][1] + dy + 128) & 127;
            const int z  = (sI4[m][2] + dz + 128) & 127;
            const int tt = (sI4[m][3] + dt + 128) & 127;
            const long base = ((((long)x*128 + y)*128 + z)*128 + tt) * 4;
            const float4 v = *(const float4*)(ts2 + base);
            float* d = &sNb[m*NB_STRIDE + j*4];
            d[0] = fmaxf(v.x, 0.f); d[1] = fmaxf(v.y, 0.f);
            d[2] = fmaxf(v.z, 0.f); d[3] = fmaxf(v.w, 0.f);
        }
    }

    // ---- stage helper: 64 points x 64 features, coalesced + ReLU ----
    const int cEnd = split*CPS + CPS;
    auto stage = [&](int c, int buf) {
        if (c >= cEnd) return;
        int seg, kseg, kglob; chunk_map(c, seg, kseg, kglob);
        float* dst = &sA[buf][0];
        #pragma unroll
        for (int i = 0; i < MT*64/256; ++i) {
            const int idx = tid + i*256;
            const int m = idx >> 6, kk = idx & 63;
            const float v = sPtr[m][seg][kseg + kk];
            dst[m*SA_STRIDE + kk] = fmaxf(v, 0.f);
        }
    };

    // ---- main K loop: 147 chunks of 64, double-buffered, WMMA fp32 ----
    v8f acc[4];
    #pragma unroll
    for (int s = 0; s < 4; ++s) acc[s] = (v8f){0.f,0.f,0.f,0.f,0.f,0.f,0.f,0.f};

    const int nCol  = wave*16 + (lane & 15);   // W_mlp row (output column)
    const int kHalf = (lane >> 4) << 1;        // lanes 16-31 carry K=2,3
    const float* Brow = Wmlp + (long)nCol * KTOT;

    const int cBeg = split*CPS;
    stage(cBeg, 0);
    for (int ii = 0; ii < CPS; ++ii) {
        const int c = cBeg + ii;
        __syncthreads();                 // stage(c) visible; compute(c-1) done
        stage(c + 1, (ii + 1) & 1);
        int seg, kseg, kglob; chunk_map(c, seg, kseg, kglob);
        const float* Bs = Brow + kglob + kHalf;
        v2f bq[16];
        #pragma unroll
        for (int k4 = 0; k4 < 16; ++k4) bq[k4] = *(const v2f*)(Bs + k4*4);
        const float* A0 = &sA[ii & 1][(lane & 15)*SA_STRIDE + kHalf];
        #pragma unroll
        for (int s = 0; s < 4; ++s) {              // 4 M sub-tiles reuse bq[]
            const float* As = A0 + s*16*SA_STRIDE;
            #pragma unroll
            for (int k4 = 0; k4 < 16; ++k4) {
                v2f a = *(const v2f*)(As + k4*4);  // ds_load_b64
                acc[s] = __builtin_amdgcn_wmma_f32_16x16x4_f32(
                             false, a, false, bq[k4], (short)0, acc[s], false, false);
            }
        }
    }
    __syncthreads();

    // ---- split 3 only: neighbor tail, 54 exact K=4 WMMAs from LDS ----
    if (split == NSPLIT - 1) {
        const float* Bs = Brow + 37632 + kHalf;
        const float* A0 = &sNb[(lane & 15)*NB_STRIDE + kHalf];
        #pragma unroll 6
        for (int j = 0; j < 54; ++j) {
            v2f b = *(const v2f*)(Bs + j*4);
            #pragma unroll
            for (int s = 0; s < 4; ++s) {
                v2f a = *(const v2f*)(A0 + s*16*NB_STRIDE + j*4);
                acc[s] = __builtin_amdgcn_wmma_f32_16x16x4_f32(
                             false, a, false, b, (short)0, acc[s], false, false);
            }
        }
    }

    // ---- write partial D tiles: P[split][point][col] ----
    // C/D layout: vgpr r, lanes 0-15 -> M=r ; lanes 16-31 -> M=r+8
    {
        const int col  = wave*16 + (lane & 15);
        const int mOff = (lane >> 4) * 8;
        #pragma unroll
        for (int s = 0; s < 4; ++s) {
            const long rowBase = (long)split*nPts + p0 + s*16 + mOff;
            #pragma unroll
            for (int r = 0; r < 8; ++r)
                P[(rowBase + r)*128 + col] = acc[s][r];
        }
    }
}

// ---- combine partials + final 132->4 projection ----
__global__ __launch_bounds__(256)
void stslt_proj_kernel(const float* __restrict__ P, const float* __restrict__ dirv,
                       const float* __restrict__ tarr, const float* __restrict__ Wproj,
                       const float* __restrict__ bproj, float* __restrict__ out,
                       int nPts)
{
    const int t = blockIdx.x*256 + threadIdx.x;
    if (t >= nPts*4) return;
    const int p = t >> 2, o = t & 3;
    const float* wp = Wproj + o*132;
    const float* p0 = P + (long)p*128;
    const long  str = (long)nPts*128;
    float s = bproj[o];
    #pragma unroll 4
    for (int n = 0; n < 128; ++n) {
        const float d = p0[n] + p0[str + n] + p0[2*str + n] + p0[3*str + n];
        s += d * wp[n];
    }
    s += dirv[p*3+0]*wp[128] + dirv[p*3+1]*wp[129] + dirv[p*3+2]*wp[130]
       + tarr[p]*wp[131];
    out[p*4 + o] = s;
}

extern "C" void kernel_launch(void* const* d_in, const int* in_sizes, int n_in,
                              void* d_out, int out_size, void* d_ws, size_t ws_size,
                              hipStream_t stream) {
    const float* pos    = (const float*)d_in[0];
    const float* dirv   = (const float*)d_in[1];
    const float* tarr   = (const float*)d_in[2];
    const float* table0 = (const float*)d_in[3];
    const float* table1 = (const float*)d_in[4];
    const float* table2 = (const float*)d_in[5];
    const float* table3 = (const float*)d_in[6];
    const float* ts1    = (const float*)d_in[7];
    const float* ts2    = (const float*)d_in[8];
    const float* Wmlp   = (const float*)d_in[9];
    const float* Wproj  = (const float*)d_in[10];
    const float* bproj  = (const float*)d_in[11];
    float* out = (float*)d_out;
    float* P   = (float*)d_ws;            // [NSPLIT][nPts][128] partials (4 MB)

    const int N = in_sizes[2];            // 2048 points
    dim3 grid(N / MT, NSPLIT);            // 32 M-tiles x 4 K-splits
    stslt_gemm_kernel<<<grid, 256, 0, stream>>>(pos, tarr, table0, table1, table2,
                                                table3, ts1, ts2, Wmlp, P, N);
    stslt_proj_kernel<<<(N*4 + 255)/256, 256, 0, stream>>>(P, dirv, tarr, Wproj,
                                                           bproj, out, N);
}